// UpSampleLayer_11579231830596
// MI455X (gfx1250) — compile-verified
//
#include <hip/hip_runtime.h>

typedef float v4f __attribute__((ext_vector_type(4)));

#define HH   128   // input H
#define WW   128   // input W
#define C4N  32    // float4 groups per pixel (128 ch / 4)
#define OHH  256
#define OWW  256
#define R_PER_BLK 16

// Depthwise transposed conv 3x3, stride 2, SAME (out = 2*in), NHWC fp32.
// One thread = one 2x2 output block x 4 channels.
__global__ __launch_bounds__(256) void dw_tconv2x_kernel(
    const float* __restrict__ x, const float* __restrict__ w,
    float* __restrict__ out)
{
    // 9 taps * 128 channels = 1152 floats = 288 float4 (4.5 KB)
    __shared__ v4f wlds[288];

    const int tid = threadIdx.x;

    // ---- CDNA5 async copy: weight tensor -> LDS (tracked by ASYNCcnt) ----
    for (int idx = tid; idx < 288; idx += 256) {
        unsigned lds_off = (unsigned)(unsigned long long)(&wlds[idx]); // low 32b of flat LDS ptr = LDS byte offset
        unsigned long long gaddr =
            (unsigned long long)w + (unsigned long long)idx * 16ull;
        asm volatile("global_load_async_to_lds_b128 %0, %1, off"
                     :: "v"(lds_off), "v"(gaddr) : "memory");
    }
    asm volatile("s_wait_asynccnt 0" ::: "memory");
    __syncthreads();

    // ---- block / thread decomposition ----
    const int b    = blockIdx.x;     // 0 .. 1023
    const int sblk = b & 15;         // 16 column tiles of 8
    const int rblk = (b >> 4) & 7;   // 8 row tiles of 16
    const int n    = b >> 7;         // 8 images

    const int wave = tid >> 5;       // 0..7  (wave-uniform column)
    const int c4   = tid & 31;       // float4 channel group, lane id
    const int s    = sblk * 8 + wave;        // input column 0..127
    const int r0   = rblk * R_PER_BLK;       // first input row

    // hoist the 9 per-channel taps into registers (w layout: (kh,kw,1,C))
    const v4f W00 = wlds[0*32 + c4], W01 = wlds[1*32 + c4], W02 = wlds[2*32 + c4];
    const v4f W10 = wlds[3*32 + c4], W11 = wlds[4*32 + c4], W12 = wlds[5*32 + c4];
    const v4f W20 = wlds[6*32 + c4], W21 = wlds[7*32 + c4], W22 = wlds[8*32 + c4];

    const v4f* __restrict__ x4 = (const v4f*)x;
    v4f* __restrict__ o4       = (v4f*)out;

    const int xbase = n * (HH * WW * C4N);
    const v4f zero  = (v4f)0.0f;
    const bool sgt0 = (s > 0);   // wave-uniform predicate

    // carried previous input row (x[r-1, s-1], x[r-1, s])
    v4f xm1_l = zero, xm1_c = zero;
    if (r0 > 0) {
        xm1_c = x4[xbase + ((r0 - 1) * WW + s) * C4N + c4];
        if (sgt0) xm1_l = x4[xbase + ((r0 - 1) * WW + (s - 1)) * C4N + c4];
    }

    for (int r = r0; r < r0 + R_PER_BLK; ++r) {
        const v4f xc_c = x4[xbase + (r * WW + s) * C4N + c4];
        const v4f xc_l = sgt0 ? x4[xbase + (r * WW + (s - 1)) * C4N + c4] : zero;

        // parity-decomposed transposed conv
        v4f o00 = xm1_l * W22 + xm1_c * W20 + xc_l * W02 + xc_c * W00;
        v4f o01 = xm1_c * W21 + xc_c * W01;
        v4f o10 = xc_l  * W12 + xc_c * W10;
        v4f o11 = xc_c  * W11;

        // write-once 256MB stream: non-temporal so x stays resident in L2
        const int ob = ((n * OHH + 2 * r) * OWW + 2 * s) * C4N + c4;
        __builtin_nontemporal_store(o00, o4 + ob);
        __builtin_nontemporal_store(o01, o4 + ob + C4N);
        __builtin_nontemporal_store(o10, o4 + ob + OWW * C4N);
        __builtin_nontemporal_store(o11, o4 + ob + OWW * C4N + C4N);

        xm1_l = xc_l;
        xm1_c = xc_c;
    }
}

extern "C" void kernel_launch(void* const* d_in, const int* in_sizes, int n_in,
                              void* d_out, int out_size, void* d_ws, size_t ws_size,
                              hipStream_t stream) {
    (void)in_sizes; (void)n_in; (void)out_size; (void)d_ws; (void)ws_size;
    const float* x = (const float*)d_in[0];
    const float* w = (const float*)d_in[1];
    float* out     = (float*)d_out;

    dim3 grid(8 * 16 * 8);   // n(8) * sblk(16) * rblk(8) = 1024 blocks
    dim3 block(256);         // 8 waves; wave = one input column, lane = 4 channels
    dw_tconv2x_kernel<<<grid, block, 0, stream>>>(x, w, out);
}